// CoherentLoss_53343493816816
// MI455X (gfx1250) — compile-verified
//
#include <hip/hip_runtime.h>
#include <math.h>

typedef float v2f __attribute__((ext_vector_type(2)));
typedef float v8f __attribute__((ext_vector_type(8)));

#define QMINc  (-6.0f)
#define QBINSc 128
#define NBINSc (128*128)          // 16384
#define Bc     4
#define Nc     131072             // 2^17
#define Mc     128
#define TOTc   (Bc*Nc)            // 524288
#define DQc    (12.0f/128.0f)     // bin width (q and p identical)
#define INVDQc (128.0f/12.0f)
#define CNORMc 0.89324408f        // (2/pi)^0.25

// ---------------------------------------------------------------------------
// Pass A: per-trajectory bins + presence bitmap + gt quadrature via WMMA f32.
// Wave layout: 16 trajectories/wave; lanes L and L+16 handle the same
// trajectory (L = lane&15) but even/odd quadrature points, matching the
// V_WMMA_F32_16X16X4_F32 A-matrix layout (lanes 0-15: K=0,1; 16-31: K=2,3).
// ---------------------------------------------------------------------------
__global__ __launch_bounds__(256) void kA(
    const float* __restrict__ q_re, const float* __restrict__ q_im,
    const float* __restrict__ p_re, const float* __restrict__ p_im,
    const float* __restrict__ spl_x, const float* __restrict__ psi_re,
    const float* __restrict__ psi_im,
    unsigned* __restrict__ presence, float2* __restrict__ gt_tab,
    float* __restrict__ partials)
{
    __shared__ v2f   btab[64 * 32];   // B-matrix table: [m-pair j][lane] -> (VGPR0,VGPR1)
    __shared__ float sx[Mc];
    __shared__ float wsc[8 * 32];     // per-wave C-matrix extraction scratch
    __shared__ float red[256];

    const int tid = threadIdx.x;

    if (tid < Mc) sx[tid] = spl_x[tid];

    // Build B table. B is 4x16 f32 (2 VGPRs): VGPRv, lane l -> row K=(l<16?0:2)+v,
    // col=l&15. Row 2m+0 = cos-part, row 2m+1 = sin-part of quadrature point m.
    // Col 0 accumulates Re(gt), col 1 accumulates Im(gt); trapezoid weight w
    // (computed from actual diff(spl_x)) is folded in. Cols 2..15 are zero.
    for (int e = tid; e < 64 * 32; e += 256) {
        int j = e >> 5;
        int l = e & 31;
        int col = l & 15;
        int kbase = (l < 16) ? 0 : 2;
        v2f val; val.x = 0.f; val.y = 0.f;
        if (col < 2) {
            #pragma unroll
            for (int v = 0; v < 2; ++v) {
                int K = kbase + v;
                int m = 2 * j + (K >> 1);
                float xm = spl_x[m];
                float wl = (m > 0)      ? (xm - spl_x[m - 1]) : 0.f;
                float wr = (m < Mc - 1) ? (spl_x[m + 1] - xm) : 0.f;
                float w  = 0.5f * (wl + wr);
                float pr = psi_re[m] * w, pi = psi_im[m] * w;
                float x;
                if (K & 1) x = (col == 0) ? pi : -pr;   // sin row
                else       x = (col == 0) ? pr :  pi;   // cos row
                if (v == 0) val.x = x; else val.y = x;
            }
        }
        btab[e] = val;
    }
    __syncthreads();

    const int lane = tid & 31;
    const int wib  = tid >> 5;
    const int half = lane >> 4;      // 0: even m, 1: odd m
    const int tl   = lane & 15;      // trajectory slot within wave
    const int waveGlobal = blockIdx.x * 8 + wib;
    const int Wtot = gridDim.x * 8;
    const int G = TOTc / 16;         // 32768 trajectory groups

    float lossLocal = 0.f;

    for (int g = waveGlobal; g < G; g += Wtot) {
        int t = g * 16 + tl;
        float qre = q_re[t], qim = q_im[t], pre = p_re[t], pim = p_im[t];
        float qf = qre - 0.5f * pim;
        float pf = 2.0f * qim + pre;
        float qb = floorf((qf - QMINc) * INVDQc);
        float pb = floorf((pf - QMINc) * INVDQc);
        int bin = (int)(qb * 128.0f + pb);
        bin = bin < 0 ? 0 : (bin > NBINSc - 1 ? NBINSc - 1 : bin);
        if (half == 0) presence[bin] = 1u;          // benign race: all write 1

        float qc = (qb + 0.5f) * DQc + QMINc;       // unclipped bin centers
        float pc = (pb + 0.5f) * DQc + QMINc;

        v8f acc = {0.f, 0.f, 0.f, 0.f, 0.f, 0.f, 0.f, 0.f};
        #pragma unroll 4
        for (int j = 0; j < 64; ++j) {
            float d = sx[2 * j + half] - qc;
            float e = __expf(-d * d);
            float s, c;
            __sincosf(pc * d, &s, &c);
            v2f a; a.x = e * c; a.y = e * s;        // A[traj, K] per ISA layout
            v2f bb = btab[j * 32 + lane];
            acc = __builtin_amdgcn_wmma_f32_16x16x4_f32(
                false, a, false, bb, (short)0, acc, false, false);
        }

        // C/D 16x16 f32 layout: lane 0/16 hold col0 rows 0-7/8-15, lanes 1/17 col1.
        float* sc = &wsc[wib * 32];
        if ((lane & 30) == 0 || (lane & 30) == 16) {
            int col = lane & 1;
            int mb = (lane >= 16) ? 8 : 0;
            #pragma unroll
            for (int r = 0; r < 8; ++r) sc[(mb + r) * 2 + col] = acc[r];
        }
        asm volatile("s_wait_dscnt 0x0" ::: "memory");  // same-wave DS RAW fence
        if (half == 0) {
            float gr = CNORMc * sc[tl * 2 + 0];
            float gi = CNORMc * sc[tl * 2 + 1];
            lossLocal += gr * gr + gi * gi;          // sum |gt|^2 over all n
            int b = t >> 17;
            int n = t & (Nc - 1);
            if (n < NBINSc) { float2 v; v.x = gr; v.y = gi; gt_tab[(b << 14) + n] = v; }
        }
    }

    red[tid] = lossLocal;
    __syncthreads();
    for (int s = 128; s > 0; s >>= 1) {
        if (tid < s) red[tid] += red[tid + s];
        __syncthreads();
    }
    if (tid == 0) partials[blockIdx.x] = red[0];
}

// ---------------------------------------------------------------------------
// Pass B: exclusive prefix sum over the 16384-bin presence bitmap.
// rank[v] = #present bins < v  ==  searchsorted(unique(bins), v).
// ---------------------------------------------------------------------------
__global__ __launch_bounds__(256) void kB(const unsigned* __restrict__ presence,
                                          unsigned* __restrict__ rank)
{
    __shared__ unsigned sums[256];
    int t = threadIdx.x;
    int base = t * 64;
    unsigned s = 0;
    for (int k = 0; k < 64; ++k) s += presence[base + k];
    sums[t] = s;
    __syncthreads();
    if (t == 0) {
        unsigned run = 0;
        for (int i = 0; i < 256; ++i) { unsigned tmp = sums[i]; sums[i] = run; run += tmp; }
    }
    __syncthreads();
    unsigned run = sums[t];
    for (int k = 0; k < 64; ++k) {
        unsigned v = presence[base + k];
        rank[base + k] = run;
        run += v;
    }
}

// ---------------------------------------------------------------------------
// Pass C: prefactor * factors scattered into compacted bins (f32 HW atomics).
// ---------------------------------------------------------------------------
__global__ __launch_bounds__(256) void kC(
    const float* __restrict__ q_re, const float* __restrict__ q_im,
    const float* __restrict__ p_re, const float* __restrict__ p_im,
    const float* __restrict__ f_re, const float* __restrict__ f_im,
    const unsigned* __restrict__ rank, float* __restrict__ V)
{
    int t = blockIdx.x * 256 + threadIdx.x;
    if (t >= TOTc) return;
    float qre = q_re[t], qim = q_im[t], pre = p_re[t], pim = p_im[t];
    float qf = qre - 0.5f * pim;
    float pf = 2.0f * qim + pre;
    float qb = floorf((qf - QMINc) * INVDQc);
    float pb = floorf((pf - QMINc) * INVDQc);
    int bin = (int)(qb * 128.0f + pb);
    bin = bin < 0 ? 0 : (bin > NBINSc - 1 ? NBINSc - 1 : bin);
    unsigned j = rank[bin];

    // pref = exp(i*p*q), p,q complex: z = p*q -> exp(i z) = e^{-Im z}(cos Re z + i sin Re z)
    float zr = pre * qre - pim * qim;
    float zi = pre * qim + pim * qre;
    float ex = __expf(-zi);
    float s, c; __sincosf(zr, &s, &c);
    float prr = ex * c, pii = ex * s;
    prr = __builtin_isnan(prr) ? 0.f : prr;
    pii = __builtin_isnan(pii) ? 0.f : pii;
    prr = fminf(100.f, fmaxf(-100.f, prr));
    pii = fminf(100.f, fmaxf(-100.f, pii));

    float fr = f_re[t], fi = f_im[t];
    float sr = prr * fr - pii * fi;
    float si = prr * fi + pii * fr;
    int b = t >> 17;
    float* dst = V + (size_t)(((unsigned)b << 14) + j) * 2;
    unsafeAtomicAdd(dst,     sr);
    unsafeAtomicAdd(dst + 1, si);
}

// ---------------------------------------------------------------------------
// Pass D: loss = sum|gt|^2 + sum_j (|V|^2 - 2 Re(conj(V)*gt_j)); out = sqrt.
// (term is automatically 0 where V==0, covering untouched compacted slots)
// ---------------------------------------------------------------------------
__global__ __launch_bounds__(256) void kD(const float* __restrict__ V,
                                          const float2* __restrict__ gt_tab,
                                          const float* __restrict__ partials,
                                          int nPartials, float* __restrict__ out)
{
    __shared__ double red[256];
    int t = threadIdx.x;
    double acc = 0.0;
    for (int i = t; i < Bc * NBINSc; i += 256) {
        float vr = V[2 * i], vi = V[2 * i + 1];
        float2 g = gt_tab[i];
        acc += (double)vr * vr + (double)vi * vi
             - 2.0 * ((double)vr * g.x + (double)vi * g.y);
    }
    for (int i = t; i < nPartials; i += 256) acc += (double)partials[i];
    red[t] = acc;
    __syncthreads();
    for (int s = 128; s > 0; s >>= 1) {
        if (t < s) red[t] += red[t + s];
        __syncthreads();
    }
    if (t == 0) out[0] = sqrtf((float)fmax(red[0], 0.0));
}

// ---------------------------------------------------------------------------
extern "C" void kernel_launch(void* const* d_in, const int* in_sizes, int n_in,
                              void* d_out, int out_size, void* d_ws, size_t ws_size,
                              hipStream_t stream)
{
    const float* q_re  = (const float*)d_in[0];
    const float* q_im  = (const float*)d_in[1];
    const float* p_re  = (const float*)d_in[2];
    const float* p_im  = (const float*)d_in[3];
    const float* f_re  = (const float*)d_in[4];
    const float* f_im  = (const float*)d_in[5];
    const float* spl_x = (const float*)d_in[6];
    const float* psi_re = (const float*)d_in[7];
    const float* psi_im = (const float*)d_in[8];

    char* ws = (char*)d_ws;
    unsigned* presence = (unsigned*)(ws + 0);         //  64 KB
    unsigned* rank     = (unsigned*)(ws + 65536);     //  64 KB
    float*    V        = (float*)   (ws + 131072);    // 512 KB (4*16384 complex)
    float2*   gt_tab   = (float2*)  (ws + 655360);    // 512 KB
    float*    partials = (float*)   (ws + 1179648);   //   4 KB

    // zero presence + rank + V each call (harness does not re-poison)
    hipMemsetAsync(ws, 0, 655360, stream);

    kA<<<1024, 256, 0, stream>>>(q_re, q_im, p_re, p_im, spl_x, psi_re, psi_im,
                                 presence, gt_tab, partials);
    kB<<<1, 256, 0, stream>>>(presence, rank);
    kC<<<TOTc / 256, 256, 0, stream>>>(q_re, q_im, p_re, p_im, f_re, f_im, rank, V);
    kD<<<1, 256, 0, stream>>>(V, gt_tab, partials, 1024, (float*)d_out);
}